// DeepSetClassifier_65532611002916
// MI455X (gfx1250) — compile-verified
//
#include <hip/hip_runtime.h>
#include <math.h>

#define BSZ 8
#define EE  8
#define NN  128
#define DIM 32

typedef __attribute__((ext_vector_type(2))) float v2f;
typedef __attribute__((ext_vector_type(8))) float v8f;

__device__ __forceinline__ float gelu_exact(float x) {
    // exact gelu: 0.5*x*(1+erf(x/sqrt(2)))   (reference uses approximate=False)
    return 0.5f * x * (1.0f + erff(x * 0.70710678118654752f));
}

__global__ __launch_bounds__(256)
void deepset_fused_kernel(const float* __restrict__ z,      // (BS,E,N,N)
                          const float* __restrict__ mask,   // (BS,N,N)
                          const float* __restrict__ phi_w,  // (DIM)
                          const float* __restrict__ phi_b,  // (DIM)
                          const float* __restrict__ wself_w,// (DIM)
                          const float* __restrict__ wself_b,// (DIM)
                          const float* __restrict__ wctx,   // (DIM,DIM) row-major [k][d]
                          const float* __restrict__ out_w,  // (DIM)
                          const float* __restrict__ out_b,  // scalar
                          float* __restrict__ out)          // (BS,E,N,N)
{
    __shared__ float sPhiW[DIM], sPhiB[DIM], sSelfW[DIM], sSelfB[DIM], sOutW[DIM];
    __shared__ float sWctx[DIM * DIM];
    __shared__ float sS[16], sM[16];
    __shared__ float sC[16][DIM];    // c tile   (16 rows x 32 dims)
    __shared__ float sCtx[16][DIM];  // ctx tile (16 rows x 32 k)

    const int t  = threadIdx.x;
    const int ng = blockIdx.x;   // 16-row group: 0..7
    const int e  = blockIdx.y;
    const int b  = blockIdx.z;

    const int n0 = ng * 16;
    const float* zb = z    + (((size_t)(b * EE + e)) * NN + n0) * NN;
    const float* mb = mask + (((size_t)b) * NN + n0) * NN;
    float*       ob = out  + (((size_t)(b * EE + e)) * NN + n0) * NN;

    // ---- stage small weights into LDS ----
    if (t < DIM) {
        sPhiW[t]  = phi_w[t];   sPhiB[t]  = phi_b[t];
        sSelfW[t] = wself_w[t]; sSelfB[t] = wself_b[t];
        sOutW[t]  = out_w[t];
    }
    for (int i = t; i < DIM * DIM; i += 256) sWctx[i] = wctx[i];
    __syncthreads();

    // ---- Phase A: per-row masked sums S[n], M[n] (coalesced, wave32 xor-reduction) ----
    const int wave = t >> 5, lane = t & 31;
    for (int r = wave; r < 16; r += 8) {
        float s = 0.f, msum = 0.f;
        #pragma unroll
        for (int j = 0; j < 4; ++j) {
            const int m = lane + 32 * j;
            const float mk = mb[r * NN + m];
            s    = fmaf(mk, zb[r * NN + m], s);
            msum += mk;
        }
        #pragma unroll
        for (int off = 16; off >= 1; off >>= 1) {
            s    += __shfl_xor(s, off, 32);
            msum += __shfl_xor(msum, off, 32);
        }
        if (lane == 0) { sS[r] = s; sM[r] = msum; }
    }
    __syncthreads();

    // ---- Phase B: c[n][d] = (phi_w[d]*S + phi_b[d]*M) / max(M,1) ----
    for (int i = t; i < 16 * DIM; i += 256) {
        const int n = i >> 5, d = i & 31;
        const float M = sM[n];
        const float inv = 1.0f / fmaxf(M, 1.0f);
        sC[n][d] = (sPhiW[d] * sS[n] + sPhiB[d] * M) * inv;
    }
    __syncthreads();

    // ---- Phase C: ctx[n][k] = sum_d c[n][d] * wctx[k][d]  via V_WMMA_F32_16X16X4_F32 ----
    // Wave 0 only (uniform branch -> EXEC all-1s inside). 8 accumulating WMMAs (K=4 each)
    // per 16-column half of the output; 16 WMMAs total.
    if (t < 32) {
        const int l    = t;
        const int mrow = l & 15;          // A row / B col / D col index
        const int koff = (l >> 4) << 1;   // K sub-offset: lanes 0-15 -> {0,1}, 16-31 -> {2,3}
        v8f acc0 = {}, acc1 = {};
        #pragma unroll
        for (int kk = 0; kk < 8; ++kk) {
            const int d0 = kk * 4 + koff;
            v2f a, b0v, b1v;
            a[0]   = sC[mrow][d0];
            a[1]   = sC[mrow][d0 + 1];
            b0v[0] = sWctx[mrow * DIM + d0];            // k = mrow      (cols 0..15)
            b0v[1] = sWctx[mrow * DIM + d0 + 1];
            b1v[0] = sWctx[(mrow + 16) * DIM + d0];     // k = mrow + 16 (cols 16..31)
            b1v[1] = sWctx[(mrow + 16) * DIM + d0 + 1];
            acc0 = __builtin_amdgcn_wmma_f32_16x16x4_f32(
                       false, a, false, b0v, (short)0, acc0, false, false);
            acc1 = __builtin_amdgcn_wmma_f32_16x16x4_f32(
                       false, a, false, b1v, (short)0, acc1, false, false);
        }
        // D layout: element v, lanes 0-15 -> row v ; lanes 16-31 -> row v+8 ; col = l&15
        const int rbase = (l >> 4) * 8;
        const int col   = l & 15;
        #pragma unroll
        for (int v = 0; v < 8; ++v) {
            sCtx[rbase + v][col]      = acc0[v];
            sCtx[rbase + v][col + 16] = acc1[v];
        }
    }
    __syncthreads();

    // ---- Phase D: out[n][m] = sum_d gelu(z*wself_w + wself_b + ctx[n][d]) * out_w[d] + out_b ----
    const float outb = out_b[0];
    const int m  = t & 127;       // coalesced along m
    const int r0 = t >> 7;        // 0 or 1
    #pragma unroll 2
    for (int i = 0; i < 8; ++i) {
        const int n = r0 + 2 * i;
        const float zv = zb[n * NN + m];
        float acc = 0.f;
        #pragma unroll
        for (int d = 0; d < DIM; ++d) {
            const float x = fmaf(zv, sSelfW[d], sSelfB[d]) + sCtx[n][d];
            acc = fmaf(gelu_exact(x), sOutW[d], acc);
        }
        ob[n * NN + m] = acc + outb;
    }
}

extern "C" void kernel_launch(void* const* d_in, const int* in_sizes, int n_in,
                              void* d_out, int out_size, void* d_ws, size_t ws_size,
                              hipStream_t stream) {
    (void)in_sizes; (void)n_in; (void)out_size; (void)d_ws; (void)ws_size;
    dim3 grid(8, EE, BSZ);   // 512 blocks x 256 threads (8 wave32s) = 4096 waves
    dim3 block(256);
    deepset_fused_kernel<<<grid, block, 0, stream>>>(
        (const float*)d_in[0],  // z_tilde
        (const float*)d_in[1],  // mask
        (const float*)d_in[2],  // phi_w
        (const float*)d_in[3],  // phi_b
        (const float*)d_in[4],  // wself_w
        (const float*)d_in[5],  // wself_b
        (const float*)d_in[6],  // wctx_w
        (const float*)d_in[7],  // out_w
        (const float*)d_in[8],  // out_b (scalar, 1-elem array)
        (float*)d_out);
}